// ConvLayer_56710748176450
// MI455X (gfx1250) — compile-verified
//
#include <hip/hip_runtime.h>
#include <hip/hip_bf16.h>

// ---- problem dimensions (match reference) ----
#define NDST   50000
#define DSELF  128
#define DNEIGH 128
#define DEDGE  32
#define DAGG   (DNEIGH + DEDGE)   // 160
#define DOUT   256

typedef float v2f __attribute__((ext_vector_type(2)));
typedef float v8f __attribute__((ext_vector_type(8)));

// ---------------------------------------------------------------------------
// Kernel 1: zero the accumulation workspace (Xagg [NDST x 160] + deg [NDST])
// ---------------------------------------------------------------------------
__global__ __launch_bounds__(256)
void zero_ws_kernel(float* __restrict__ ws, long long n) {
    long long i = (long long)blockIdx.x * blockDim.x + threadIdx.x;
    long long stride = (long long)gridDim.x * blockDim.x;
    for (; i < n; i += stride) ws[i] = 0.0f;
}

// ---------------------------------------------------------------------------
// Kernel 2: edge scatter. One wave32 per edge.
//   lanes:  4 neigh floats each (float4 load) + 1 edge float each
//   lane 0: degree count
// Atomic adds are fire-and-forget (non-returning -> global_atomic_add_f32,
// tracked on STOREcnt). h_neigh (25.6MB) and Xagg (32MB) are L2-resident.
// ---------------------------------------------------------------------------
__global__ __launch_bounds__(256)
void scatter_kernel(const float* __restrict__ h_neigh,
                    const float* __restrict__ edge_feats,
                    const int*   __restrict__ src,
                    const int*   __restrict__ dst,
                    float*       __restrict__ Xagg,   // [NDST][DAGG]
                    float*       __restrict__ deg,    // [NDST]
                    int n_edges) {
    int e    = (int)(((long long)blockIdx.x * blockDim.x + threadIdx.x) >> 5);
    int lane = threadIdx.x & 31;
    if (e >= n_edges) return;                 // wave-uniform exit

    int s = src[e];
    int d = dst[e];

    const float* hn = h_neigh + (size_t)s * DNEIGH;
    float*       xd = Xagg    + (size_t)d * DAGG;

    // 128 neighbor floats -> 4 per lane
    float4 v = *(const float4*)(hn + lane * 4);
    atomicAdd(xd + lane * 4 + 0, v.x);
    atomicAdd(xd + lane * 4 + 1, v.y);
    atomicAdd(xd + lane * 4 + 2, v.z);
    atomicAdd(xd + lane * 4 + 3, v.w);

    // 32 edge floats -> 1 per lane
    float ev = edge_feats[(size_t)e * DEDGE + lane];
    atomicAdd(xd + DNEIGH + lane, ev);

    if (lane == 0) atomicAdd(deg + d, 1.0f);
}

// ---------------------------------------------------------------------------
// Kernel 3: mean-divide:  Xagg[r][c] /= max(deg[r], 1)
// ---------------------------------------------------------------------------
__global__ __launch_bounds__(256)
void scale_kernel(float* __restrict__ Xagg, const float* __restrict__ deg) {
    long long i = (long long)blockIdx.x * blockDim.x + threadIdx.x;
    if (i >= (long long)NDST * DAGG) return;
    int r = (int)(i / DAGG);
    float dg = deg[r];
    float dn = (dg == 0.0f) ? 1.0f : dg;
    Xagg[i] = Xagg[i] / dn;
}

// ---------------------------------------------------------------------------
// Kernel 4: fused fp32-WMMA GEMM + ReLU + row L2 normalize.
//   z[m][n] = relu( sum_k Xc[m][k] * Wc[n][k] ),  K = 288 (128 self + 160 agg)
//   out[m][:] = z[m][:] / ||z[m][:]||2
// Block = 512 threads = 16 waves; block owns 16 rows (M), wave w owns
// N-tile [16w, 16w+16). Each wave runs 72x V_WMMA_F32_16X16X4_F32.
// A fragment (ISA 7.12.2, 32-bit 16x4): lane&15 = M row, VGPR v holds
// K = k + v + 2*(lane>>4)  -> one float2 load at Xrow + k + 2h.
// B fragment mirrors it with lane&15 = N (weight row n).
// ---------------------------------------------------------------------------
__global__ __launch_bounds__(512)
void gemm_norm_kernel(const float* __restrict__ h_self,   // [NDST][128]
                      const float* __restrict__ Xagg,     // [NDST][160]
                      const float* __restrict__ W_self,   // [256][128]
                      const float* __restrict__ W_neigh,  // [256][160]
                      float*       __restrict__ out) {    // [NDST][256]
    __shared__ float tile[16 * DOUT];   // 16KB z slab
    __shared__ float rnorm[16];

    const int tid  = threadIdx.x;
    const int wave = tid >> 5;          // 0..15 -> N tile
    const int lane = tid & 31;
    const int h    = lane >> 4;         // lane half: K offset 0 / 2
    const int l16  = lane & 15;

    const int m0 = blockIdx.x * 16;     // 3125 blocks * 16 = 50000 rows exactly
    const int n0 = wave * 16;
    const int m  = m0 + l16;            // A row for this lane
    const int n  = n0 + l16;            // weight row (= B column) for this lane

    const float* xs = h_self  + (size_t)m * DSELF;
    const float* xa = Xagg    + (size_t)m * DAGG;
    const float* ws = W_self  + (size_t)n * DSELF;
    const float* wn = W_neigh + (size_t)n * DAGG;

    v8f c = {0.f, 0.f, 0.f, 0.f, 0.f, 0.f, 0.f, 0.f};

    // K = 0..127 : h_self x W_self
    #pragma unroll 8
    for (int k = 0; k < DSELF; k += 4) {
        v2f a = *(const v2f*)(xs + k + 2 * h);
        v2f b = *(const v2f*)(ws + k + 2 * h);
        c = __builtin_amdgcn_wmma_f32_16x16x4_f32(false, a, false, b,
                                                  (short)0, c, false, false);
    }
    // K = 128..287 : [neigh_agg|edge_agg] x W_neigh
    #pragma unroll 8
    for (int k = 0; k < DAGG; k += 4) {
        v2f a = *(const v2f*)(xa + k + 2 * h);
        v2f b = *(const v2f*)(wn + k + 2 * h);
        c = __builtin_amdgcn_wmma_f32_16x16x4_f32(false, a, false, b,
                                                  (short)0, c, false, false);
    }

    // ReLU, spill 16x16 tile to the 16x256 LDS slab.
    // C layout: lanes 0-15 -> M = j, N = lane; lanes 16-31 -> M = j+8.
    #pragma unroll
    for (int j = 0; j < 8; ++j) {
        float v = c[j];
        v = v > 0.0f ? v : 0.0f;
        tile[(j + 8 * h) * DOUT + n0 + l16] = v;
    }
    __syncthreads();

    // Wave w reduces row w's 256 squares (8/lane + shuffle tree).
    {
        float s = 0.0f;
        #pragma unroll
        for (int cc = 0; cc < DOUT; cc += 32) {
            float v = tile[wave * DOUT + cc + lane];
            s += v * v;
        }
        for (int off = 16; off > 0; off >>= 1) s += __shfl_xor(s, off, 32);
        if (lane == 0) {
            float nrm = sqrtf(s);
            rnorm[wave] = (nrm == 0.0f) ? 1.0f : nrm;
        }
    }
    __syncthreads();

    // Normalized store: 4096 elements / 512 threads = 8 each.
    for (int i = tid; i < 16 * DOUT; i += 512) {
        int r = i >> 8;
        out[(size_t)(m0 + r) * DOUT + (i & (DOUT - 1))] = tile[i] / rnorm[r];
    }
}

// ---------------------------------------------------------------------------
// launcher
// inputs: 0 h_neigh, 1 h_self, 2 edge_feats, 3 src, 4 dst, 5 W_self, 6 W_neigh
// ---------------------------------------------------------------------------
extern "C" void kernel_launch(void* const* d_in, const int* in_sizes, int n_in,
                              void* d_out, int out_size, void* d_ws, size_t ws_size,
                              hipStream_t stream) {
    const float* h_neigh    = (const float*)d_in[0];
    const float* h_self     = (const float*)d_in[1];
    const float* edge_feats = (const float*)d_in[2];
    const int*   src        = (const int*)d_in[3];
    const int*   dst        = (const int*)d_in[4];
    const float* W_self     = (const float*)d_in[5];
    const float* W_neigh    = (const float*)d_in[6];
    float*       out        = (float*)d_out;

    const int n_edges = in_sizes[3];

    float* Xagg = (float*)d_ws;                        // NDST*160 floats (32MB)
    float* deg  = Xagg + (size_t)NDST * DAGG;          // NDST floats

    // 1) zero accumulators
    {
        long long n = (long long)NDST * DAGG + NDST;
        int blocks = (int)((n + 255) / 256);
        if (blocks > 65535) blocks = 65535;            // grid-stride
        zero_ws_kernel<<<blocks, 256, 0, stream>>>((float*)d_ws, n);
    }
    // 2) edge scatter (one wave per edge, 8 waves/block)
    {
        int blocks = (n_edges + 7) / 8;
        scatter_kernel<<<blocks, 256, 0, stream>>>(h_neigh, edge_feats, src, dst,
                                                   Xagg, deg, n_edges);
    }
    // 3) mean divide
    {
        long long n = (long long)NDST * DAGG;
        int blocks = (int)((n + 255) / 256);
        scale_kernel<<<blocks, 256, 0, stream>>>(Xagg, deg);
    }
    // 4) fused WMMA GEMM + ReLU + L2 normalize
    {
        int blocks = NDST / 16;                        // 3125, exact
        gemm_norm_kernel<<<blocks, 512, 0, stream>>>(h_self, Xagg, W_self, W_neigh, out);
    }
}